// GDMLTorchPredict_10179072491528
// MI455X (gfx1250) — compile-verified
//
#include <hip/hip_runtime.h>
#include <hip/hip_bf16.h>
#include <math.h>

// ---------------------------------------------------------------------------
// GDML predict for MI455X (gfx1250). All four GEMM-shaped reductions run on
// V_WMMA_F32_16X16X4_F32 (full fp32 precision, wave32). D=210 is zero-padded
// to 224 in staged workspace copies so both WMMA K-loops are branch-free and
// every fragment load is a single 8-byte global_load_b64.
// ---------------------------------------------------------------------------

typedef float v2f __attribute__((ext_vector_type(2)));
typedef float v8f __attribute__((ext_vector_type(8)));

#define MGEO   128
#define NATOM  21
#define DPAIR  210      // 21*20/2
#define DPAD   224      // padded: covers K-pad (212) and N-tile pad (14*16)
#define TTRAIN 4096

__device__ __forceinline__ float QF()    { return 0.22360679775f; } // sqrt(5)/10
__device__ __forceinline__ float INVQF() { return 4.47213595500f; } // 10/sqrt(5)
__device__ __forceinline__ float K5()    { return 5.0f / 300.0f; }  // 5/(3*SIG^2)

__device__ __forceinline__ v8f wmma_f32(v2f a, v2f b, v8f c) {
  // D = A(16x4) * B(4x16) + C(16x16), all fp32, wave32
  return __builtin_amdgcn_wmma_f32_16x16x4_f32(
      /*neg_a=*/false, a, /*neg_b=*/false, b,
      /*c_mod=*/(short)0, c, /*reuse_a=*/false, /*reuse_b=*/false);
}

// --- workspace layout (float offsets), total ~19.1 MB ----------------------
#define OFF_XSP    0                            // xs padded [128*224]
#define OFF_XN2    (OFF_XSP + MGEO*DPAD)        // [128]
#define OFF_SX     (OFF_XN2 + MGEO)             // [4096]
#define OFF_SXA    (OFF_SX + TTRAIN)            // [4096]
#define OFF_SUMW1  (OFF_SXA + TTRAIN)           // [128]
#define OFF_ESACC  (OFF_SUMW1 + MGEO)           // [128]
#define OFF_XTP    (OFF_ESACC + MGEO)           // xs_train padded [4096*224]
#define OFF_JXP    (OFF_XTP + TTRAIN*DPAD)      // Jx_alphas padded [4096*224]
#define OFF_W1     (OFF_JXP + TTRAIN*DPAD)      // W1 [M,T] = [128*4096]
#define OFF_E1     (OFF_W1 + MGEO*TTRAIN)       // E1 [M,T]
#define OFF_FSX    (OFF_E1 + MGEO*TTRAIN)       // Fs_x [128*210]
#define OFF_XTT    (OFF_FSX + MGEO*DPAIR)       // xs_train^T padded [224*4096]
#define OFF_JXT    (OFF_XTT + DPAD*TTRAIN)      // Jx_alphas^T padded [224*4096]

// ---------------------------------------------------------------------------
// Kernel A: xs padded descriptors, xn2[m]; zero per-m accumulators.
// ---------------------------------------------------------------------------
__global__ __launch_bounds__(256)
void kA_descriptors(const float* __restrict__ Rs, float* __restrict__ ws) {
  float* xsp   = ws + OFF_XSP;
  float* xn2   = ws + OFF_XN2;
  float* sumW1 = ws + OFF_SUMW1;
  float* esacc = ws + OFF_ESACC;

  const int m = blockIdx.x;
  const int p = threadIdx.x;
  __shared__ float Rm[NATOM * 3];
  __shared__ float red[256];
  if (p < NATOM * 3) Rm[p] = Rs[m * NATOM * 3 + p];
  __syncthreads();

  float contrib = 0.0f;
  if (p < DPAIR) {
    // row index of strict-lower-tri pair p (numpy tril order)
    int i = (int)((1.0f + sqrtf(1.0f + 8.0f * (float)p)) * 0.5f);
    if (i < 1) i = 1;
    while (i * (i - 1) / 2 > p) --i;
    while ((i + 1) * i / 2 <= p) ++i;
    int j = p - i * (i - 1) / 2;
    float dx = Rm[i*3+0] - Rm[j*3+0];
    float dy = Rm[i*3+1] - Rm[j*3+1];
    float dz = Rm[i*3+2] - Rm[j*3+2];
    float inv = 1.0f / sqrtf(dx*dx + dy*dy + dz*dz);
    xsp[m * DPAD + p] = inv;
    contrib = inv * inv;
  } else if (p < DPAD) {
    xsp[m * DPAD + p] = 0.0f;                    // zero pad tail
  }
  red[p] = contrib;
  __syncthreads();
  for (int s = 128; s > 0; s >>= 1) {
    if (p < s) red[p] += red[p + s];
    __syncthreads();
  }
  if (p == 0) {
    xn2[m]   = red[0];
    sumW1[m] = 0.0f;
    esacc[m] = 0.0f;
  }
}

// ---------------------------------------------------------------------------
// Kernel B: stage zero-padded row-major + transposed copies of xs_train and
// Jx_alphas, and reduce per-row scalars sx[t], sxa[t]. One block per t.
// ---------------------------------------------------------------------------
__global__ __launch_bounds__(256)
void kB_stage(const float* __restrict__ xt, const float* __restrict__ jxa,
              float* __restrict__ ws) {
  float* sx  = ws + OFF_SX;
  float* sxa = ws + OFF_SXA;
  float* xtp = ws + OFF_XTP;
  float* jxp = ws + OFF_JXP;
  float* xtt = ws + OFF_XTT;
  float* jxt = ws + OFF_JXT;

  const int t = blockIdx.x;
  const int d = threadIdx.x;
  __shared__ float r1[256], r2[256];
  float a = 0.0f, b = 0.0f;
  if (d < DPAD) {
    const bool ok = (d < DPAIR);
    float vx = ok ? xt [t * DPAIR + d] : 0.0f;
    float vj = ok ? jxa[t * DPAIR + d] : 0.0f;
    xtp[t * DPAD + d] = vx;
    jxp[t * DPAD + d] = vj;
    xtt[d * TTRAIN + t] = vx;
    jxt[d * TTRAIN + t] = vj;
    a = vx * vx;
    b = vx * vj;
  }
  r1[d] = a; r2[d] = b;
  __syncthreads();
  for (int s = 128; s > 0; s >>= 1) {
    if (d < s) { r1[d] += r1[d + s]; r2[d] += r2[d + s]; }
    __syncthreads();
  }
  if (d == 0) { sx[t] = r1[0]; sxa[t] = r2[0]; }
}

// ---------------------------------------------------------------------------
// Phase 1: G1 = xs_train . xs^T, G2 = Jx . xs^T via WMMA f32 16x16x4 over the
// padded K=212; branch-free loop, b64 fragment loads. Output tiles are [t x m];
// fused epilogue -> W1, E1 stored [M,T], plus row-sum atomics.
// One wave per 16x16 tile: 256 tTiles * 8 mTiles / 8 waves = 256 blocks.
// ---------------------------------------------------------------------------
__global__ __launch_bounds__(256)
void kP1_kernelweights(float* __restrict__ ws) {
  const float* xsp = ws + OFF_XSP;
  const float* xn2 = ws + OFF_XN2;
  const float* sx  = ws + OFF_SX;
  const float* sxa = ws + OFF_SXA;
  const float* xtp = ws + OFF_XTP;
  const float* jxp = ws + OFF_JXP;
  float* W1buf = ws + OFF_W1;
  float* E1buf = ws + OFF_E1;
  float* sumW1 = ws + OFF_SUMW1;
  float* esacc = ws + OFF_ESACC;

  const int lane  = threadIdx.x & 31;
  const int wave  = threadIdx.x >> 5;
  const int tile  = blockIdx.x * 8 + wave;       // 0..2047
  const int tTile = tile >> 3;                   // 0..255
  const int mTile = tile & 7;                    // 0..7
  const int t0 = tTile * 16, m0 = mTile * 16;
  const int r    = lane & 15;
  const int koff = (lane >> 4) << 1;             // lanes 0-15 -> K 0/1, 16-31 -> K 2/3

  v8f acc1, acc2;
  for (int q = 0; q < 8; ++q) { acc1[q] = 0.0f; acc2[q] = 0.0f; }

  const float* a1row = xtp + (t0 + r) * DPAD + koff;  // A rows: contiguous K
  const float* a2row = jxp + (t0 + r) * DPAD + koff;
  const float* brow  = xsp + (m0 + r) * DPAD + koff;  // B col n = lane&15

  #pragma unroll 4
  for (int d0 = 0; d0 < 212; d0 += 4) {          // K padded to 212 (53 steps)
    v2f a1 = *(const v2f*)(a1row + d0);
    v2f a2 = *(const v2f*)(a2row + d0);
    v2f b  = *(const v2f*)(brow  + d0);
    acc1 = wmma_f32(a1, b, acc1);                // G1 tile [t x m]
    acc2 = wmma_f32(a2, b, acc2);                // G2 tile [t x m]
  }

  // epilogue: element (row = v + 8*hi -> t, col = lane&15 -> m)
  const int m    = m0 + r;
  const int hi   = (lane >> 4) * 8;
  const float xm = xn2[m];
  float pw = 0.0f, pe = 0.0f;
  #pragma unroll
  for (int v = 0; v < 8; ++v) {
    const int t = t0 + v + hi;
    float d2  = xm - 2.0f * acc1[v] + sx[t];
    float xd  = QF() * sqrtf(fmaxf(d2, 0.0f));
    float ex  = K5() * __expf(-xd);
    float dv  = QF() * (acc2[v] - sxa[t]);
    float w1  = ex * dv;
    float e1  = ex * (1.0f + xd);
    W1buf[m * TTRAIN + t] = w1;                  // [M,T] for coalesced phase-2 A
    E1buf[m * TTRAIN + t] = e1;
    pw += w1;
    pe += e1 * dv;
  }
  atomicAdd(&sumW1[m], pw);
  atomicAdd(&esacc[m], pe);
}

// ---------------------------------------------------------------------------
// Phase 2: F1raw = W1 . xs_train, F2 = E1 . Jx_alphas via WMMA (K = 4096).
// A frags: b64 from [M,T] buffers; B frags: b64 from transposed padded copies.
// Branch-free hot loop: 4x global_load_b64 + 2x wmma per step.
// Grid: 8 mTiles * 14 dTiles = 112 waves -> 14 blocks of 8 waves.
// ---------------------------------------------------------------------------
__global__ __launch_bounds__(256)
void kP2_forces_desc(float* __restrict__ ws) {
  const float* xsp   = ws + OFF_XSP;
  const float* W1buf = ws + OFF_W1;
  const float* E1buf = ws + OFF_E1;
  const float* sumW1 = ws + OFF_SUMW1;
  const float* xtt   = ws + OFF_XTT;
  const float* jxt   = ws + OFF_JXT;
  float* fsx = ws + OFF_FSX;

  const int lane  = threadIdx.x & 31;
  const int wave  = threadIdx.x >> 5;
  const int tile  = blockIdx.x * 8 + wave;       // 0..111
  const int mTile = tile & 7;                    // 0..7
  const int dTile = tile >> 3;                   // 0..13
  const int m0 = mTile * 16, d0 = dTile * 16;
  const int r    = lane & 15;
  const int koff = (lane >> 4) << 1;

  v8f accF1, accF2;
  for (int q = 0; q < 8; ++q) { accF1[q] = 0.0f; accF2[q] = 0.0f; }

  const int dcol = d0 + r;                       // 0..223, always valid (padded)
  const float* aWrow = W1buf + (m0 + r) * TTRAIN + koff;  // A row m, contiguous k
  const float* aErow = E1buf + (m0 + r) * TTRAIN + koff;
  const float* bXrow = xtt + dcol * TTRAIN + koff;        // B col d, contiguous k
  const float* bJrow = jxt + dcol * TTRAIN + koff;

  #pragma unroll 4
  for (int t0 = 0; t0 < TTRAIN; t0 += 4) {
    v2f aW = *(const v2f*)(aWrow + t0);
    v2f aE = *(const v2f*)(aErow + t0);
    v2f bX = *(const v2f*)(bXrow + t0);
    v2f bJ = *(const v2f*)(bJrow + t0);
    accF1 = wmma_f32(aW, bX, accF1);             // sum_t W1 * xs_train
    accF2 = wmma_f32(aE, bJ, accF2);             // sum_t E1 * Jx_alphas
  }

  if (dcol < DPAIR) {
    const int hi = (lane >> 4) * 8;
    #pragma unroll
    for (int v = 0; v < 8; ++v) {
      const int m = m0 + v + hi;
      // Fs_x = Q*(xs[m,d]*sumW1 - F1raw) - F2   (STD = 1)
      float f = QF() * (xsp[m * DPAD + dcol] * sumW1[m] - accF1[v]) - accF2[v];
      fsx[m * DPAIR + dcol] = f;
    }
  }
}

// ---------------------------------------------------------------------------
// Phase 3: per-geometry force assembly + energy.
// out[0:128] = Es ; out[128:8192] = Fs [128,21,3]
// ---------------------------------------------------------------------------
__global__ __launch_bounds__(64)
void kP3_assemble(const float* __restrict__ Rs, const float* __restrict__ ws,
                  float* __restrict__ out) {
  const float* fsxAll = ws + OFF_FSX;
  const float* esacc  = ws + OFF_ESACC;
  const int m   = blockIdx.x;
  const int tid = threadIdx.x;

  __shared__ float Rm[NATOM * 3];
  __shared__ float frow[DPAIR];
  for (int q = tid; q < NATOM * 3; q += 64) Rm[q] = Rs[m * NATOM * 3 + q];
  for (int q = tid; q < DPAIR; q += 64)     frow[q] = fsxAll[m * DPAIR + q];
  __syncthreads();

  if (tid == 0) out[m] = esacc[m] * INVQF();     // C = 0, STD = 1

  if (tid < NATOM * 3) {
    const int b = tid / 3;
    const int c = tid % 3;
    const float bx = Rm[b*3+0], by = Rm[b*3+1], bz = Rm[b*3+2];
    float acc = 0.0f;
    for (int a = 0; a < NATOM; ++a) {
      if (a == b) continue;
      float dx = Rm[a*3+0] - bx;
      float dy = Rm[a*3+1] - by;
      float dz = Rm[a*3+2] - bz;
      float dist2 = dx*dx + dy*dy + dz*dz;
      float dist  = sqrtf(dist2);
      int p = (a > b) ? (a*(a-1)/2 + b) : (b*(b-1)/2 + a);
      float val = frow[p] / (dist2 * dist);      // full * inv_d3
      acc += val * ((c == 0) ? dx : (c == 1) ? dy : dz);
    }
    out[MGEO + m * (NATOM * 3) + tid] = acc;
  }
}

// ---------------------------------------------------------------------------
extern "C" void kernel_launch(void* const* d_in, const int* in_sizes, int n_in,
                              void* d_out, int out_size, void* d_ws, size_t ws_size,
                              hipStream_t stream) {
  const float* Rs  = (const float*)d_in[0];   // [128,21,3]
  const float* xt  = (const float*)d_in[1];   // [4096,210]
  const float* jxa = (const float*)d_in[2];   // [4096,210]
  float* out = (float*)d_out;                 // [128] Es ++ [128,21,3] Fs
  float* ws  = (float*)d_ws;

  kA_descriptors<<<MGEO, 256, 0, stream>>>(Rs, ws);
  kB_stage<<<TTRAIN, 256, 0, stream>>>(xt, jxa, ws);
  // 256 tTiles x 8 mTiles = 2048 wave-tiles, 8 waves/block
  kP1_kernelweights<<<256, 256, 0, stream>>>(ws);
  // 8 mTiles x 14 dTiles = 112 wave-tiles, 8 waves/block
  kP2_forces_desc<<<14, 256, 0, stream>>>(ws);
  kP3_assemble<<<MGEO, 64, 0, stream>>>(Rs, ws, out);
}